// Model_61125974557070
// MI455X (gfx1250) — compile-verified
//
#include <hip/hip_runtime.h>
#include <hip/hip_bf16.h>
#include <math.h>
#include <stdint.h>

#define B_   4
#define L_   512
#define DM_  256
#define H_   4
#define TD_  8
#define TW_  16
#define D_   64          // DM/H
#define PHW_ 48          // 3*TW per head
#define ROWS_ (B_ * L_)  // 2048
#define UCOLS_ (3 * H_ * TW_)  // 192
#define PHS_ 52          // LDS row stride for u/v tiles: 208B -> 16B aligned rows

// Flip to 0 if the assembler rejects the async-to-LDS mnemonics.
#define USE_ASYNC_LDS 1

typedef __attribute__((ext_vector_type(2))) float v2f;
typedef __attribute__((ext_vector_type(4))) float v4f;
typedef __attribute__((ext_vector_type(8))) float v8f;

static __device__ __forceinline__ v8f wmma_f32_16x16x4(v2f a, v2f b, v8f c) {
  // 8 args: (neg_a, A, neg_b, B, c_mod, C, reuse_a, reuse_b)
  return __builtin_amdgcn_wmma_f32_16x16x4_f32(false, a, false, b, (short)0, c,
                                               false, false);
}

static __device__ __forceinline__ v8f v8f_zero() {
  v8f r;
#pragma unroll
  for (int i = 0; i < 8; ++i) r[i] = 0.f;
  return r;
}

// 16-byte global -> LDS stage. Async (ASYNCcnt-tracked, bypasses VGPRs) when
// available; plain b128 load+ds_store otherwise. Both require 16B alignment.
static __device__ __forceinline__ void stage16B(const float* g, float* l) {
#if USE_ASYNC_LDS
  asm volatile("global_load_async_to_lds_b128 %0, %1, off"
               :
               : "v"((uint32_t)(uintptr_t)l), "v"((uint64_t)(uintptr_t)g)
               : "memory");
#else
  *(v4f*)l = *(const v4f*)g;
#endif
}

static __device__ __forceinline__ void stage_wait() {
#if USE_ASYNC_LDS
  asm volatile("s_wait_asynccnt 0" ::: "memory");
#endif
}

// ---------------------------------------------------------------------------
// Kernel 1: fused QKV projections. C[m,n] = sum_k In[m,k] * W[n,k]
// One wave per 16x16 output tile, K-loop of 64 x v_wmma_f32_16x16x4_f32.
// Output written directly in [B, H, L, D] layout.
// ---------------------------------------------------------------------------
__global__ __launch_bounds__(128) void hawkes_proj_kernel(
    const float* __restrict__ Q, const float* __restrict__ K,
    const float* __restrict__ V, const float* __restrict__ WQ,
    const float* __restrict__ WK, const float* __restrict__ WV,
    float* __restrict__ Q0, float* __restrict__ K0, float* __restrict__ V0) {
  const int lane = threadIdx.x & 31;
  const int wave = threadIdx.x >> 5;
  const int mt = blockIdx.y * 4 + wave;  // 0..127  (M = 2048)
  const int nt = blockIdx.x;             // 0..15   (N = 256)
  const int p = blockIdx.z;              // 0:Q 1:K 2:V

  const float* In = (p == 0) ? Q : (p == 1) ? K : V;
  const float* W = (p == 0) ? WQ : (p == 1) ? WK : WV;
  float* Out = (p == 0) ? Q0 : (p == 1) ? K0 : V0;

  const int m0 = mt * 16, n0 = nt * 16;
  const int am = lane & 15;          // row within tile (A) / col (B)
  const int kb2 = (lane >> 4) << 1;  // K sub-offset {0,2}

  v8f c = v8f_zero();
  const float* arow = In + (m0 + am) * DM_ + kb2;
  const float* brow = W + (n0 + am) * DM_ + kb2;  // B[k][n] = W[n][k]
#pragma unroll 8
  for (int k0 = 0; k0 < DM_; k0 += 4) {
    v2f a = *(const v2f*)(arow + k0);
    v2f b = *(const v2f*)(brow + k0);
    c = wmma_f32_16x16x4(a, b, c);
  }

  // C/D layout: vgpr v -> row v + 8*(lane>>4); col = lane&15
  const int rg = (lane >> 4) << 3;
#pragma unroll
  for (int v = 0; v < 8; ++v) {
    const int m = m0 + v + rg;
    const int n = n0 + am;
    const int bb = m >> 9, l = m & (L_ - 1);
    const int h = n >> 6, d = n & (D_ - 1);
    Out[(((bb * H_ + h) * L_) + l) * D_ + d] = c[v];
  }
}

// ---------------------------------------------------------------------------
// Kernel 2: factorized time-MLP features.
//   delta[q,k,t] = t_Q[q,t] - t_K[k,t] is rank-structured, so layer-0 of the
//   per-pair MLP factorizes:  z[q,k,c] = u[q,c] - v[k,c]
//   u[row, c] = sum_t t_Q[row,t] * pw0[ph,t,w] + pb0[c]   (c = ph*16 + w)
//   v[row, c] = sum_t t_K[row,t] * pw0[ph,t,w]
// ---------------------------------------------------------------------------
__global__ __launch_bounds__(UCOLS_) void hawkes_timefeat_kernel(
    const float* __restrict__ tQ, const float* __restrict__ tK,
    const float* __restrict__ pw0, const float* __restrict__ pb0,
    float* __restrict__ U, float* __restrict__ Vt) {
  const int row = blockIdx.x;  // 0..2047
  const int c = threadIdx.x;   // 0..191 = (p*H + h)*16 + w
  const int ph = c >> 4, w = c & 15;
  float u = pb0[c];
  float v = 0.f;
#pragma unroll
  for (int t = 0; t < TD_; ++t) {
    const float wt = pw0[(ph * TD_ + t) * TW_ + w];
    u += tQ[row * TD_ + t] * wt;
    v += tK[row * TD_ + t] * wt;
  }
  U[row * UCOLS_ + c] = u;
  Vt[row * UCOLS_ + c] = v;
}

// ---------------------------------------------------------------------------
// Kernel 3: flash-style Hawkes attention. One wave per (b, h, q-tile).
//   S = phiQ*phiK*(Q0 K0^T)/sqrt(D), mask k>=q, online softmax,
//   O += softmax(S)*phiV @ V0.   QK^T and PV use fp32 WMMA; time features are
//   staged with async-to-LDS; phi uses packed f32 math from registers.
// ---------------------------------------------------------------------------
__global__ __launch_bounds__(32) void hawkes_attn_kernel(
    const float* __restrict__ Q0, const float* __restrict__ K0,
    const float* __restrict__ V0, const float* __restrict__ U,
    const float* __restrict__ Vt, const float* __restrict__ pw1,
    const float* __restrict__ pb1, const int* __restrict__ maskp,
    float* __restrict__ out) {
  const int lane = threadIdx.x;
  const int qt = blockIdx.x;  // 0..31
  const int bh = blockIdx.y;  // 0..15
  const int b = bh >> 2;
  const int h = bh & 3;
  const int qb = qt * 16;

  __shared__ __align__(16) float lds_u[16][PHS_];
  __shared__ __align__(16) float lds_v[16][PHS_];
  __shared__ __align__(16) float lds_p[16][18];  // even stride: aligned v2f

  // ---- stage q-tile time features: 16 rows x 3 segments x 4 x 16B chunks
  //      packed LDS col = p*16 + w  <->  global col = p*64 + h*16 + w
#pragma unroll
  for (int it = 0; it < 6; ++it) {
    const int c = lane + 32 * it;  // 0..191
    const int r = c / 12, seg = (c % 12) >> 2, j4 = (c & 3) << 2;
    stage16B(U + (b * L_ + qb + r) * UCOLS_ + seg * 64 + h * TW_ + j4,
             &lds_u[r][seg * TW_ + j4]);
  }

  const int am = lane & 15;
  const int kb2 = (lane >> 4) << 1;
  const int rg = (lane >> 4) << 3;
  const int n = am;

  // layer-1 weights/biases for this head, held in registers (uniform loads)
  v2f w1r[24];
#pragma unroll
  for (int p = 0; p < 3; ++p)
#pragma unroll
    for (int w2 = 0; w2 < 8; ++w2)
      w1r[p * 8 + w2] = *(const v2f*)(pw1 + (p * H_ + h) * TW_ + 2 * w2);
  const float b1q = pb1[0 * H_ + h];
  const float b1k = pb1[1 * H_ + h];
  const float b1v = pb1[2 * H_ + h];

  // preload Q A-fragments for all 16 K-steps (d = 0..63 in chunks of 4)
  const float* q0row = Q0 + (bh * L_ + qb + am) * D_ + kb2;
  v2f qa[16];
#pragma unroll
  for (int t = 0; t < 16; ++t) qa[t] = *(const v2f*)(q0row + 4 * t);

  float rm[8], rl[8];
  v8f acc[4];
#pragma unroll
  for (int v = 0; v < 8; ++v) { rm[v] = -INFINITY; rl[v] = 0.f; }
#pragma unroll
  for (int nf = 0; nf < 4; ++nf) acc[nf] = v8f_zero();

  const int causal = *maskp;
  const int kstart = causal ? qb : 0;
  const v2f zero2 = {0.f, 0.f};

  for (int kb = kstart; kb < L_; kb += 16) {
    if (kb + 16 < L_) {
      __builtin_prefetch(K0 + (bh * L_ + kb + 16) * D_, 0, 0);
      __builtin_prefetch(V0 + (bh * L_ + kb + 16) * D_, 0, 0);
    }
    // ---- stage k-tile time features (async; ASYNCcnt-tracked)
#pragma unroll
    for (int it = 0; it < 6; ++it) {
      const int c = lane + 32 * it;
      const int r = c / 12, seg = (c % 12) >> 2, j4 = (c & 3) << 2;
      stage16B(Vt + (b * L_ + kb + r) * UCOLS_ + seg * 64 + h * TW_ + j4,
               &lds_v[r][seg * TW_ + j4]);
    }
    __syncthreads();

    // ---- S = Q0 K0^T (16x16 tile) via 16 fp32 WMMAs
    v8f s = v8f_zero();
    const float* krow = K0 + (bh * L_ + kb + am) * D_ + kb2;
#pragma unroll
    for (int t = 0; t < 16; ++t) {
      v2f bf = *(const v2f*)(krow + 4 * t);  // B[d][n] = K0[kb+n][d]
      s = wmma_f32_16x16x4(qa[t], bf, s);
    }

    stage_wait();  // async v-tile (and first-iter u-tile) landed in LDS

    // this lane's k-column features -> registers (reused by all 8 rows)
    v2f vv[24];
    {
      const v2f* vrow = (const v2f*)&lds_v[n][0];
#pragma unroll
      for (int i = 0; i < 24; ++i) vv[i] = vrow[i];
    }

    // ---- per-element phi_p = b1 + sum_w relu(u-v)*w1 (packed f32 math)
    float fvv[8];
#pragma unroll
    for (int v = 0; v < 8; ++v) {
      const int m = v + rg;
      const v2f* urow = (const v2f*)&lds_u[m][0];
      v2f aq = zero2, ak = zero2, av = zero2;
#pragma unroll
      for (int w2 = 0; w2 < 8; ++w2) {
        v2f d0 = urow[w2] - vv[w2];
        v2f d1 = urow[8 + w2] - vv[8 + w2];
        v2f d2 = urow[16 + w2] - vv[16 + w2];
        d0 = __builtin_elementwise_max(d0, zero2);
        d1 = __builtin_elementwise_max(d1, zero2);
        d2 = __builtin_elementwise_max(d2, zero2);
        aq += d0 * w1r[w2];
        ak += d1 * w1r[8 + w2];
        av += d2 * w1r[16 + w2];
      }
      const float pq = b1q + aq[0] + aq[1];
      const float pk = b1k + ak[0] + ak[1];
      fvv[v] = b1v + av[0] + av[1];
      float sv = s[v] * 0.125f * pq * pk;  // 1/sqrt(D=64)
      if (causal && (kb + n) < (qb + m)) sv = -INFINITY;
      s[v] = sv;
    }

    // ---- online softmax per row (row spread over 16 lanes of a half-wave)
#pragma unroll
    for (int v = 0; v < 8; ++v) {
      float tm = s[v];
      tm = fmaxf(tm, __shfl_xor(tm, 1));
      tm = fmaxf(tm, __shfl_xor(tm, 2));
      tm = fmaxf(tm, __shfl_xor(tm, 4));
      tm = fmaxf(tm, __shfl_xor(tm, 8));
      const float mnew = fmaxf(rm[v], tm);
      const float alpha = __expf(rm[v] - mnew);
      float pe = __expf(s[v] - mnew);
      float rs = pe;
      rs += __shfl_xor(rs, 1);
      rs += __shfl_xor(rs, 2);
      rs += __shfl_xor(rs, 4);
      rs += __shfl_xor(rs, 8);
      rl[v] = rl[v] * alpha + rs;
      rm[v] = mnew;
#pragma unroll
      for (int nf = 0; nf < 4; ++nf) acc[nf][v] *= alpha;
      lds_p[v + rg][n] = pe * fvv[v];  // fold phiV into P
    }
    __syncthreads();

    // ---- O += P @ V0-tile  (16x16 x 16x64) via 16 fp32 WMMAs
    const float* vbase = V0 + (bh * L_ + kb) * D_;
#pragma unroll
    for (int j = 0; j < 4; ++j) {
      v2f pa = *(const v2f*)(&lds_p[am][4 * j + kb2]);  // A-frag reload
#pragma unroll
      for (int nf = 0; nf < 4; ++nf) {
        const float* vp = vbase + (4 * j + kb2) * D_ + nf * 16 + n;
        v2f bf;
        bf[0] = vp[0];
        bf[1] = vp[D_];
        acc[nf] = wmma_f32_16x16x4(pa, bf, acc[nf]);
      }
    }
    __syncthreads();
  }

  // ---- normalize + store to [B, L, H*D]
#pragma unroll
  for (int nf = 0; nf < 4; ++nf) {
#pragma unroll
    for (int v = 0; v < 8; ++v) {
      const int m = v + rg;
      out[(b * L_ + qb + m) * DM_ + h * D_ + nf * 16 + n] =
          acc[nf][v] / rl[v];
    }
  }
}

// ---------------------------------------------------------------------------
extern "C" void kernel_launch(void* const* d_in, const int* in_sizes, int n_in,
                              void* d_out, int out_size, void* d_ws,
                              size_t ws_size, hipStream_t stream) {
  (void)in_sizes; (void)n_in; (void)out_size; (void)ws_size;
  const float* Q = (const float*)d_in[0];
  const float* K = (const float*)d_in[1];
  const float* V = (const float*)d_in[2];
  const float* tQ = (const float*)d_in[3];
  const float* tK = (const float*)d_in[4];
  const float* WQ = (const float*)d_in[5];
  const float* WK = (const float*)d_in[6];
  const float* WV = (const float*)d_in[7];
  const float* pw0 = (const float*)d_in[8];
  const float* pb0 = (const float*)d_in[9];
  const float* pw1 = (const float*)d_in[10];
  const float* pb1 = (const float*)d_in[11];
  const int* maskp = (const int*)d_in[12];

  float* ws = (float*)d_ws;
  const size_t projN = (size_t)B_ * H_ * L_ * D_;  // 524288
  float* Q0 = ws;
  float* K0 = Q0 + projN;
  float* V0 = K0 + projN;
  float* U = V0 + projN;  // 2048*192
  float* Vt = U + (size_t)ROWS_ * UCOLS_;

  hawkes_proj_kernel<<<dim3(DM_ / 16, ROWS_ / 64, 3), dim3(128), 0, stream>>>(
      Q, K, V, WQ, WK, WV, Q0, K0, V0);
  hawkes_timefeat_kernel<<<dim3(ROWS_), dim3(UCOLS_), 0, stream>>>(
      tQ, tK, pw0, pb0, U, Vt);
  hawkes_attn_kernel<<<dim3(L_ / 16, B_ * H_), dim3(32), 0, stream>>>(
      Q0, K0, V0, U, Vt, pw1, pb1, maskp, (float*)d_out);
}